// Linear_4844723110442
// MI455X (gfx1250) — compile-verified
//
#include <hip/hip_runtime.h>

typedef __attribute__((ext_vector_type(2))) float v2f;
typedef __attribute__((ext_vector_type(4))) float v4f;
typedef __attribute__((ext_vector_type(8))) float v8f;

#define IN_F   4096
#define OUT_F  4096
#define HROWS  4096
#define KDIM   256   // TILE_H: reduced K dimension
#define NBLK   16    // 4096 / 256 column-blocks summed

// ---------------------------------------------------------------------------
// Reduction: out[r][k] = sum_{v<16} in[r][v*256 + k]
// One float4 of output per thread; fully coalesced reads, non-temporal
// (inputs are read exactly once -> don't pollute L2, which must keep xs/wsT).
// Used for both  x -> xs  and  weight -> wsT  (identical access pattern).
// ---------------------------------------------------------------------------
__global__ __launch_bounds__(256) void reduce_blocks_f32(
    const float* __restrict__ in, float* __restrict__ out) {
  int tid = blockIdx.x * 256 + threadIdx.x;   // one v4f of out per thread
  int r   = tid >> 6;                         // 64 float4-chunks per row
  int c4  = (tid & 63) << 2;                  // k offset within [0,256)
  const float* row = in + (size_t)r * IN_F + c4;
  v4f acc = {0.f, 0.f, 0.f, 0.f};
#pragma unroll
  for (int v = 0; v < NBLK; ++v) {
    acc += __builtin_nontemporal_load((const v4f*)(row + v * KDIM));
  }
  *(v4f*)(out + (size_t)r * KDIM + c4) = acc;   // regular store: stays in cache
}

// ---------------------------------------------------------------------------
// GEMM: out[m][n] = sum_k xs[m][k] * wsT[n][k] + bias[n]
//   M = 4096, N = 4096, K = 256, all f32, via V_WMMA_F32_16X16X4_F32.
// A-fragment layout (ISA 7.12.2, 32-bit A 16x4): lanes 0-15 hold K=k,k+1 of
// row M=lane; lanes 16-31 hold K=k+2,k+3 of row M=lane-16.  B mirrors this
// with N across lanes.  Both xs and wsT are K-contiguous, so each fragment
// is a single float2 load per lane.
// Wave computes a 4x4 grid of 16x16 tiles (64x64 out): 8 loads feed 16 WMMAs
// per K-step (16 FLOP per L2 byte).  Block = 8 waves (2x4) -> 128(M) x 256(N).
// ---------------------------------------------------------------------------
__device__ inline v8f wmma_f32_k4(v2f a, v2f b, v8f c) {
  return __builtin_amdgcn_wmma_f32_16x16x4_f32(
      /*neg_a=*/false, a, /*neg_b=*/false, b,
      /*c_mod=*/(short)0, c, /*reuse_a=*/false, /*reuse_b=*/false);
}

__global__ __launch_bounds__(256) void gemm_wmma_f32(
    const float* __restrict__ xs,    // (4096 x 256), K-contiguous
    const float* __restrict__ wsT,   // (4096 x 256), K-contiguous
    const float* __restrict__ bias,  // (4096)
    float* __restrict__ out) {       // (4096 x 4096)
  const int lane = threadIdx.x & 31;
  const int wave = threadIdx.x >> 5;       // 0..7
  const int waveM = wave >> 2;             // 0..1
  const int waveN = wave & 3;              // 0..3
  const int half = lane >> 4;              // 0: K=k,k+1   1: K=k+2,k+3
  const int l    = lane & 15;              // row within tile (A) / col (B)

  const int m_base = blockIdx.y * 128 + waveM * 64;
  const int n_base = blockIdx.x * 256 + waveN * 64;

  // Per-lane K-contiguous base pointers for the 4 A-tiles and 4 B-tiles.
  const float* pa[4];
  const float* pb[4];
#pragma unroll
  for (int i = 0; i < 4; ++i) {
    pa[i] = xs  + (size_t)(m_base + 16 * i + l) * KDIM + half * 2;
    pb[i] = wsT + (size_t)(n_base + 16 * i + l) * KDIM + half * 2;
  }

  v8f acc[4][4] = {};

#pragma unroll 4
  for (int k0 = 0; k0 < KDIM; k0 += 4) {
    v2f a[4], b[4];
#pragma unroll
    for (int i = 0; i < 4; ++i) {
      a[i] = *(const v2f*)(pa[i] + k0);
      b[i] = *(const v2f*)(pb[i] + k0);
    }
#pragma unroll
    for (int i = 0; i < 4; ++i)
#pragma unroll
      for (int j = 0; j < 4; ++j)
        acc[i][j] = wmma_f32_k4(a[i], b[j], acc[i][j]);
  }

  // C/D layout: VGPR r, lanes 0-15 -> (M=r, N=l); lanes 16-31 -> (M=r+8, N=l)
  int   ncol[4];
  float bv[4];
#pragma unroll
  for (int j = 0; j < 4; ++j) {
    ncol[j] = n_base + 16 * j + l;
    bv[j]   = bias[ncol[j]];
  }

#pragma unroll
  for (int i = 0; i < 4; ++i) {
#pragma unroll
    for (int r = 0; r < 8; ++r) {
      const int m = m_base + 16 * i + r + half * 8;
      float* orow = out + (size_t)m * OUT_F;
#pragma unroll
      for (int j = 0; j < 4; ++j)
        __builtin_nontemporal_store(acc[i][j][r] + bv[j], orow + ncol[j]);
    }
  }
}

// ---------------------------------------------------------------------------
extern "C" void kernel_launch(void* const* d_in, const int* in_sizes, int n_in,
                              void* d_out, int out_size, void* d_ws, size_t ws_size,
                              hipStream_t stream) {
  const float* x      = (const float*)d_in[0];  // (4096, 4096)
  const float* weight = (const float*)d_in[1];  // (4096, 4096)
  const float* bias   = (const float*)d_in[2];  // (4096)
  float* out = (float*)d_out;                   // (4096, 4096)

  float* xs  = (float*)d_ws;                    // (4096, 256) = 4 MB
  float* wsT = xs + (size_t)HROWS * KDIM;       // (4096, 256) = 4 MB

  // Block-sum reductions (HBM-bound part: 128 MB of reads).
  const int red_blocks = (HROWS * (KDIM / 4)) / 256;  // 1024
  reduce_blocks_f32<<<red_blocks, 256, 0, stream>>>(x, xs);
  reduce_blocks_f32<<<red_blocks, 256, 0, stream>>>(weight, wsT);

  // WMMA GEMM (+bias): operands live in L2 (8 MB), writes 64 MB (NT stores).
  dim3 grid(OUT_F / 256, HROWS / 128);
  gemm_wmma_f32<<<grid, 256, 0, stream>>>(xs, wsT, bias, out);
}